// VNEdgeConvMS_60189671686870
// MI455X (gfx1250) — compile-verified
//
#include <hip/hip_runtime.h>
#include <math.h>

typedef __attribute__((ext_vector_type(2))) float v2f;
typedef __attribute__((ext_vector_type(8))) float v8f;

#define Bsz   8
#define Csz   64
#define Nsz   2048
#define Osz   64
#define KMAX  16
#define NM    (3*Nsz)          // 6144 columns per batch (v*N+n)
#define Rrows 256              // [Wf1; Wf2-Wf1; Wd1; Wd2-Wd1]
#define QSZ   ((size_t)Bsz*NM*Rrows)   // 12,582,912 floats (~50MB)

// ---------------------------------------------------------------------------
// prep: build Wbig (256x64) and zero the reduction buffers
// ---------------------------------------------------------------------------
__global__ __launch_bounds__(256) void prep_kernel(const float* __restrict__ Wf,
                                                   const float* __restrict__ Wd,
                                                   float* __restrict__ wbig,
                                                   float* __restrict__ stats,
                                                   float* __restrict__ zsum) {
    int g = blockIdx.x * 256 + threadIdx.x;          // 0..16383
    int r = g >> 6, c = g & 63;
    float v;
    if (r < 64)        v = Wf[r*128 + c];
    else if (r < 128)  { int rr = r-64;  v = Wf[rr*128 + 64 + c] - Wf[rr*128 + c]; }
    else if (r < 192)  { int rr = r-128; v = Wd[rr*128 + c]; }
    else               { int rr = r-192; v = Wd[rr*128 + 64 + c] - Wd[rr*128 + c]; }
    wbig[g] = v;
    if (g < 256) stats[g] = 0.0f;
    if (g < 128) zsum[g]  = 0.0f;
}

// ---------------------------------------------------------------------------
// fp32 WMMA GEMM: Q[b][m][r] = sum_c Wbig[r][c] * feat[b][c][m]
// block = 128 threads (4 wave32), each wave owns one 16-wide m-tile and
// sweeps all 16 r-tiles. Wbig staged in LDS (64KB), A-frags via ds_load_b64.
// ---------------------------------------------------------------------------
__global__ __launch_bounds__(128) void gemm_kernel(const float* __restrict__ feat,
                                                   const float* __restrict__ wbig,
                                                   float* __restrict__ Q) {
    __shared__ float lA[Rrows * 64];                 // 64 KB of the 320 KB LDS
    const int tid = threadIdx.x;

    // cooperative 128-bit LDS fill of Wbig
    const float4* src = (const float4*)wbig;
    float4*       dst = (float4*)lA;
#pragma unroll
    for (int i = 0; i < 32; ++i) dst[tid + i*128] = src[tid + i*128];
    __syncthreads();

    const int lane  = tid & 31;
    const int wave  = tid >> 5;
    const int b     = blockIdx.y;
    const int m0    = (blockIdx.x*4 + wave) * 16;
    const int mcol  = m0 + (lane & 15);
    const int khalf = (lane >> 4) * 2;               // K sub-pair per half-wave

    // preload all 16 B fragments (4x16 f32 each): B[k][m] = feat[b][k][m]
    v2f bf[16];
    const float* fb = feat + (size_t)b*Csz*NM + mcol;
#pragma unroll
    for (int kk = 0; kk < 16; ++kk) {
        int c = kk*4 + khalf;
        v2f t;
        t.x = fb[(size_t)c * NM];
        t.y = fb[(size_t)(c+1) * NM];
        bf[kk] = t;
    }

    float* qb = Q + ((size_t)b*NM + mcol) * Rrows;
    const int Mrow = lane & 15;

#pragma unroll 1
    for (int rt = 0; rt < 16; ++rt) {
        const int r0 = rt * 16;
        v8f acc = {};
        const float* la = lA + (r0 + Mrow)*64 + khalf;   // A[M][k] from LDS
#pragma unroll
        for (int kk = 0; kk < 16; ++kk) {
            v2f a = *(const v2f*)(la + kk*4);            // ds_load_b64
            acc = __builtin_amdgcn_wmma_f32_16x16x4_f32(
                    false, a, false, bf[kk], (short)0, acc, false, false);
        }
        // C/D layout: VGPR j, lanes0-15 -> M=j, lanes16-31 -> M=j+8
        const int rbase = r0 + 8*(lane >> 4);
        float4 s0 = {acc[0], acc[1], acc[2], acc[3]};
        float4 s1 = {acc[4], acc[5], acc[6], acc[7]};
        *(float4*)(qb + rbase)     = s0;
        *(float4*)(qb + rbase + 4) = s1;
    }
}

// ---------------------------------------------------------------------------
// pass 1: per-channel BN statistics of ||p|| for both k-branches (k=8 prefix)
// thread = (b, n, o); LDS tree reduce then atomics.
// ---------------------------------------------------------------------------
__global__ __launch_bounds__(256) void norm_stats_kernel(const float* __restrict__ Q,
                                                         const int* __restrict__ idx,
                                                         float* __restrict__ stats) {
    __shared__ float red[256];
    const int g = blockIdx.x*256 + threadIdx.x;
    const int o = g & 63, n = (g >> 6) & (Nsz-1), b = g >> 17;
    const size_t qb = (size_t)b * NM * Rrows;

    float p2v[3];
#pragma unroll
    for (int v = 0; v < 3; ++v)
        p2v[v] = Q[qb + (size_t)(v*Nsz + n)*Rrows + 64 + o];

    const int* ip = idx + ((size_t)b*Nsz + n)*KMAX;
    float s8 = 0, q8 = 0, s16 = 0, q16 = 0;
    for (int kk = 0; kk < KMAX; ++kk) {
        const int id = ip[kk];
        float pp = 0;
#pragma unroll
        for (int v = 0; v < 3; ++v) {
            float pv = Q[qb + (size_t)(v*Nsz + id)*Rrows + o] + p2v[v];
            pp += pv*pv;
        }
        float nrm = sqrtf(pp) + 1e-6f;
        s16 += nrm; q16 += nrm*nrm;
        if (kk < 8) { s8 += nrm; q8 += nrm*nrm; }
    }
    float vals[4] = {s8, q8, s16, q16};
    for (int i = 0; i < 4; ++i) {
        red[threadIdx.x] = vals[i];
        __syncthreads();
        if (threadIdx.x < 128) red[threadIdx.x] += red[threadIdx.x + 128];
        __syncthreads();
        if (threadIdx.x < 64) {
            red[threadIdx.x] += red[threadIdx.x + 64];        // all share o == tid
            atomicAdd(&stats[i*64 + threadIdx.x], red[threadIdx.x]);
        }
        __syncthreads();
    }
}

// stats -> per-branch affine: norm_bn = s*norm + t
__global__ void bn_coef_kernel(const float* __restrict__ stats,
                               const float* __restrict__ gamma,
                               const float* __restrict__ beta,
                               float* __restrict__ coef) {
    int o = threadIdx.x;
    if (o >= 64) return;
    const float c8 = (float)((size_t)Bsz*Nsz*8);
    const float c16 = (float)((size_t)Bsz*Nsz*16);
    float m8  = stats[o]       / c8,  v8  = stats[64+o]  / c8  - m8*m8;
    float m16 = stats[128+o]   / c16, v16 = stats[192+o] / c16 - m16*m16;
    float sc8  = gamma[o] * rsqrtf(v8  + 1e-5f);
    float sc16 = gamma[o] * rsqrtf(v16 + 1e-5f);
    coef[o]       = sc8;  coef[64+o]  = beta[o] - m8*sc8;
    coef[128+o]   = sc16; coef[192+o] = beta[o] - m16*sc16;
}

// ---------------------------------------------------------------------------
// pass 2: gather + vector-LeakyReLU for both branches, mean over k, avg of 2
// ---------------------------------------------------------------------------
__global__ __launch_bounds__(256) void edge_kernel(const float* __restrict__ Q,
                                                   const int* __restrict__ idx,
                                                   const float* __restrict__ coef,
                                                   float* __restrict__ out) {
    const int g = blockIdx.x*256 + threadIdx.x;
    const int o = g & 63, n = (g >> 6) & (Nsz-1), b = g >> 17;
    const float s8 = coef[o], t8 = coef[64+o], s16 = coef[128+o], t16 = coef[192+o];
    const size_t qb = (size_t)b * NM * Rrows;

    float p2v[3], d2v[3];
#pragma unroll
    for (int v = 0; v < 3; ++v) {
        const size_t col = qb + (size_t)(v*Nsz + n)*Rrows;
        p2v[v] = Q[col + 64  + o];
        d2v[v] = Q[col + 192 + o];
    }

    const int* ip = idx + ((size_t)b*Nsz + n)*KMAX;
    float acc0 = 0, acc1 = 0, acc2 = 0;
    for (int kk = 0; kk < KMAX; ++kk) {
        const int id = ip[kk];
        if (kk + 1 < KMAX) {                               // global_prefetch_b8
            int nid = ip[kk+1];
            __builtin_prefetch(&Q[qb + (size_t)nid*Rrows + o], 0, 0);
        }
        float p[3], d[3], pp = 0, pd = 0, dd = 0;
#pragma unroll
        for (int v = 0; v < 3; ++v) {
            const size_t col = qb + (size_t)(v*Nsz + id)*Rrows;
            float pv = Q[col + o]       + p2v[v];
            float dv = Q[col + 128 + o] + d2v[v];
            p[v] = pv; d[v] = dv;
            pp += pv*pv; pd += pv*dv; dd += dv*dv;
        }
        const float nrm    = sqrtf(pp) + 1e-6f;
        const float inv_dd = 0.9f / (dd + 1e-6f);
        {   // branch k=16, weight = (1/16)*(1/2)
            float al = s16 + t16 / nrm;
            float dotb = al * pd;
            float cf = (dotb >= 0.0f) ? 0.0f : dotb * inv_dd;
            acc0 += 0.03125f * (al*p[0] - cf*d[0]);
            acc1 += 0.03125f * (al*p[1] - cf*d[1]);
            acc2 += 0.03125f * (al*p[2] - cf*d[2]);
        }
        if (kk < 8) {   // branch k=8, weight = (1/8)*(1/2)
            float al = s8 + t8 / nrm;
            float dotb = al * pd;
            float cf = (dotb >= 0.0f) ? 0.0f : dotb * inv_dd;
            acc0 += 0.0625f * (al*p[0] - cf*d[0]);
            acc1 += 0.0625f * (al*p[1] - cf*d[1]);
            acc2 += 0.0625f * (al*p[2] - cf*d[2]);
        }
    }
    const size_t ob = (((size_t)b*Osz + o)*3)*Nsz + n;
    out[ob]        = acc0;
    out[ob + Nsz]  = acc1;
    out[ob + 2*Nsz]= acc2;
}

// ---------------------------------------------------------------------------
// ZCA: per-batch moments (3 sums + 6 second moments)
// ---------------------------------------------------------------------------
__global__ __launch_bounds__(256) void zca_stats_kernel(const float* __restrict__ x,
                                                        float* __restrict__ zsum) {
    __shared__ float red[256];
    const int g = blockIdx.x*256 + threadIdx.x;
    const int n = g & (Nsz-1), c = (g >> 11) & 63, b = g >> 17;
    const size_t base = (((size_t)b*Csz + c)*3)*Nsz + n;
    const float x0 = x[base], x1 = x[base + Nsz], x2 = x[base + 2*Nsz];
    float s[9] = {x0, x1, x2, x0*x0, x0*x1, x0*x2, x1*x1, x1*x2, x2*x2};
    for (int i = 0; i < 9; ++i) {
        red[threadIdx.x] = s[i];
        __syncthreads();
        for (int off = 128; off > 0; off >>= 1) {
            if (threadIdx.x < off) red[threadIdx.x] += red[threadIdx.x + off];
            __syncthreads();
        }
        if (threadIdx.x == 0) atomicAdd(&zsum[b*16 + i], red[0]);
        __syncthreads();
    }
}

// closed-form symmetric 3x3 eig + matrix function A^(-1/2) via Lagrange
__global__ void zca_solve_kernel(const float* __restrict__ zsum,
                                 float* __restrict__ zw) {
    const int b = threadIdx.x;
    if (b >= Bsz) return;
    const float Mf = (float)(Csz * Nsz);
    const float* s = zsum + b*16;
    const float mu0 = s[0]/Mf, mu1 = s[1]/Mf, mu2 = s[2]/Mf;
    const float inv = 1.0f / (Mf + 1e-6f);
    float a00 = (s[3] - Mf*mu0*mu0)*inv + 1e-5f;
    float a01 = (s[4] - Mf*mu0*mu1)*inv;
    float a02 = (s[5] - Mf*mu0*mu2)*inv;
    float a11 = (s[6] - Mf*mu1*mu1)*inv + 1e-5f;
    float a12 = (s[7] - Mf*mu1*mu2)*inv;
    float a22 = (s[8] - Mf*mu2*mu2)*inv + 1e-5f;

    float p1 = a01*a01 + a02*a02 + a12*a12;
    float q  = (a00 + a11 + a22) / 3.0f;
    float p2 = (a00-q)*(a00-q) + (a11-q)*(a11-q) + (a22-q)*(a22-q) + 2.0f*p1;

    float w00, w01, w02, w11, w12, w22;
    if (p2 < 1e-20f) {
        float f = rsqrtf(fmaxf(q, 1e-5f));
        w00 = f; w11 = f; w22 = f; w01 = 0; w02 = 0; w12 = 0;
    } else {
        float p  = sqrtf(p2 / 6.0f);
        float ipv = 1.0f / p;
        float b00 = (a00-q)*ipv, b11 = (a11-q)*ipv, b22 = (a22-q)*ipv;
        float b01 = a01*ipv, b02 = a02*ipv, b12 = a12*ipv;
        float detB = b00*(b11*b22 - b12*b12) - b01*(b01*b22 - b12*b02)
                   + b02*(b01*b12 - b11*b02);
        float r = fminf(1.0f, fmaxf(-1.0f, detB*0.5f));
        float phi = acosf(r) / 3.0f;
        float l0 = q + 2.0f*p*cosf(phi);
        float l2 = q + 2.0f*p*cosf(phi + 2.0943951023931953f);
        float l1 = 3.0f*q - l0 - l2;

        // A^2
        float c00 = a00*a00 + a01*a01 + a02*a02;
        float c01 = a00*a01 + a01*a11 + a02*a12;
        float c02 = a00*a02 + a01*a12 + a02*a22;
        float c11 = a01*a01 + a11*a11 + a12*a12;
        float c12 = a01*a02 + a11*a12 + a12*a22;
        float c22 = a02*a02 + a12*a12 + a22*a22;

        float ls[3] = {l0, l1, l2};
        w00 = w01 = w02 = w11 = w12 = w22 = 0.0f;
        for (int i = 0; i < 3; ++i) {
            float li = ls[i], lj = ls[(i+1)%3], lk = ls[(i+2)%3];
            float den = (li - lj) * (li - lk);
            if (fabsf(den) < 1e-12f) den = (den < 0.0f) ? -1e-12f : 1e-12f;
            float sc  = rsqrtf(fmaxf(li, 1e-5f)) / den;
            float sjk = lj + lk, pjk = lj * lk;
            w00 += sc * (c00 - sjk*a00 + pjk);
            w01 += sc * (c01 - sjk*a01);
            w02 += sc * (c02 - sjk*a02);
            w11 += sc * (c11 - sjk*a11 + pjk);
            w12 += sc * (c12 - sjk*a12);
            w22 += sc * (c22 - sjk*a22 + pjk);
        }
    }
    float* o = zw + b*16;
    o[0]=w00; o[1]=w01; o[2]=w02;
    o[3]=w01; o[4]=w11; o[5]=w12;
    o[6]=w02; o[7]=w12; o[8]=w22;
    o[9]=mu0; o[10]=mu1; o[11]=mu2;
}

// apply whitening in place: y = gamma[c] * W @ (x - mu)
__global__ __launch_bounds__(256) void zca_apply_kernel(float* __restrict__ x,
                                                        const float* __restrict__ zw,
                                                        const float* __restrict__ gz) {
    const int g = blockIdx.x*256 + threadIdx.x;
    const int n = g & (Nsz-1), c = (g >> 11) & 63, b = g >> 17;
    const float* w = zw + b*16;
    const size_t base = (((size_t)b*Csz + c)*3)*Nsz + n;
    const float x0 = x[base]         - w[9];
    const float x1 = x[base + Nsz]   - w[10];
    const float x2 = x[base + 2*Nsz] - w[11];
    const float gm = gz[c];
    x[base]          = (w[0]*x0 + w[1]*x1 + w[2]*x2) * gm;
    x[base + Nsz]    = (w[3]*x0 + w[4]*x1 + w[5]*x2) * gm;
    x[base + 2*Nsz]  = (w[6]*x0 + w[7]*x1 + w[8]*x2) * gm;
}

// ---------------------------------------------------------------------------
extern "C" void kernel_launch(void* const* d_in, const int* in_sizes, int n_in,
                              void* d_out, int out_size, void* d_ws, size_t ws_size,
                              hipStream_t stream) {
    const float* feat = (const float*)d_in[0];   // (8,64,3,2048)
    const float* Wf   = (const float*)d_in[1];   // (64,128)
    const float* Wd   = (const float*)d_in[2];   // (64,128)
    const float* bng  = (const float*)d_in[3];   // (64,)
    const float* bnb  = (const float*)d_in[4];   // (64,)
    const float* zg   = (const float*)d_in[5];   // (64,)
    const int*   idx  = (const int*)d_in[6];     // (8,2048,16)
    float* out = (float*)d_out;                  // (8,64,3,2048)

    float* ws    = (float*)d_ws;
    float* Q     = ws;                 // QSZ floats (~50MB)
    float* wbig  = ws + QSZ;           // 16384
    float* stats = wbig + 16384;       // 256
    float* coef  = stats + 256;        // 256
    float* zsum  = coef + 256;         // 128
    float* zw    = zsum + 128;         // 128

    prep_kernel<<<64, 256, 0, stream>>>(Wf, Wd, wbig, stats, zsum);
    gemm_kernel<<<dim3(NM/16/4, Bsz), 128, 0, stream>>>(feat, wbig, Q);
    norm_stats_kernel<<<4096, 256, 0, stream>>>(Q, idx, stats);
    bn_coef_kernel<<<1, 64, 0, stream>>>(stats, bng, bnb, coef);
    edge_kernel<<<4096, 256, 0, stream>>>(Q, idx, coef, out);
    zca_stats_kernel<<<4096, 256, 0, stream>>>(out, zsum);
    zca_solve_kernel<<<1, 32, 0, stream>>>(zsum, zw);
    zca_apply_kernel<<<4096, 256, 0, stream>>>(out, zw, zg);
}